// GCN_58128087384148
// MI455X (gfx1250) — compile-verified
//
#include <hip/hip_runtime.h>

typedef __attribute__((ext_vector_type(2))) float v2f;
typedef __attribute__((ext_vector_type(8))) float v8f;

#define NN      100000
#define DINF    128
#define DHIDF   128
#define DOUTF   64
#define NEDGE   1600000

// ---------------------------------------------------------------------------
// zero fill, float4 (global_store_b128), grid-stride
// n must be a multiple of 4 (all our buffers are).
// ---------------------------------------------------------------------------
__global__ void zero_f32v4(float4* __restrict__ p, long n4) {
    long i = (long)blockIdx.x * blockDim.x + threadIdx.x;
    long st = (long)gridDim.x * blockDim.x;
    float4 z = make_float4(0.f, 0.f, 0.f, 0.f);
    for (; i < n4; i += st) p[i] = z;
}

// ---------------------------------------------------------------------------
// C[M x N] = act(H)[M x K] * W[N x K]^T   via V_WMMA_F32_16X16X4_F32
// One wave computes one 16x16 tile of C. blockDim.x = 32 * (N/16),
// gridDim.x = M/16. KC is a compile-time constant -> full unroll (KC/4 WMMAs).
// RELU_IN applies ReLU to H fragments on load (fuses the activation of the
// previous layer into this GEMM).
// A-frag (16x4 f32): lanes 0-15 rows m0..m0+15 at k,k+1; lanes 16-31 same
// rows at k+2,k+3. B[k][n] = W[n][k], mirrored layout. C: VGPR i -> row
// m0+i+(lane>=16)*8, col n0+(lane&15).
// ---------------------------------------------------------------------------
template <int KC, bool RELU_IN>
__global__ void gemm_wmma_f32(const float* __restrict__ H,
                              const float* __restrict__ W,
                              float* __restrict__ C,
                              int N) {
    const int lane  = threadIdx.x & 31;
    const int wave  = threadIdx.x >> 5;       // N-tile index
    const int m0    = blockIdx.x * 16;
    const int n0    = wave * 16;
    const int half  = lane >> 4;              // 0: lanes 0-15, 1: lanes 16-31
    const int l15   = lane & 15;
    const int kbase = half * 2;               // upper half-wave covers k+2,k+3

    const float* hrow = H + (long)(m0 + l15) * KC;
    const float* wrow = W + (long)(n0 + l15) * KC;

    v8f acc = {};
    #pragma unroll
    for (int k = 0; k < KC; k += 4) {
        v2f a, b;
        a.x = hrow[k + kbase];
        a.y = hrow[k + kbase + 1];
        if (RELU_IN) {
            a.x = a.x > 0.f ? a.x : 0.f;
            a.y = a.y > 0.f ? a.y : 0.f;
        }
        b.x = wrow[k + kbase];
        b.y = wrow[k + kbase + 1];
        acc = __builtin_amdgcn_wmma_f32_16x16x4_f32(
            /*neg_a=*/false, a, /*neg_b=*/false, b,
            /*c_mod=*/(short)0, acc, /*reuse_a=*/false, /*reuse_b=*/false);
    }

    float* cp = C + (long)(m0 + half * 8) * N + n0 + l15;
    #pragma unroll
    for (int i = 0; i < 8; ++i) cp[(long)i * N] = acc[i];
}

// ---------------------------------------------------------------------------
// COO SpMM scatter: y[row[e]*D + d] += val[e] * h[col[e]*D + d]
// (1 << SHIFT) consecutive threads share one edge; each thread handles one
// float4 chunk (coalesced 512B gather of h[col]) then 4 hardware fp32
// atomics (global_atomic_add_f32) into the L2-resident accumulator.
// UNIFORM_E: when SHIFT==5 an edge spans exactly one wave32, so force the
// edge index scalar with readfirstlane -> edge metadata comes in via SMEM.
// ---------------------------------------------------------------------------
template <int SHIFT, bool UNIFORM_E>
__global__ void spmm_atomic(const int*   __restrict__ erow,
                            const int*   __restrict__ ecol,
                            const float* __restrict__ eval,
                            const float* __restrict__ h,
                            float*       __restrict__ y,
                            long totalChunks) {
    constexpr int D = 4 << SHIFT;
    long g = (long)blockIdx.x * blockDim.x + threadIdx.x;
    if (g >= totalChunks) return;
    int e;
    if (UNIFORM_E) {
        e = __builtin_amdgcn_readfirstlane((int)(g >> SHIFT));
    } else {
        e = (int)(g >> SHIFT);
    }
    int q = (int)(g & ((1 << SHIFT) - 1));

    int   r = erow[e];
    int   c = ecol[e];
    float v = eval[e];

    const float4 hv = *reinterpret_cast<const float4*>(h + (long)c * D + q * 4);
    float* yp = y + (long)r * D + q * 4;
    unsafeAtomicAdd(yp + 0, v * hv.x);
    unsafeAtomicAdd(yp + 1, v * hv.y);
    unsafeAtomicAdd(yp + 2, v * hv.z);
    unsafeAtomicAdd(yp + 3, v * hv.w);
}

// ---------------------------------------------------------------------------
// residual merge: dst = relu(dst) + relu(res)
// (res holds the pre-activation layer-0 accumulator; relu(res) == h1)
// ---------------------------------------------------------------------------
__global__ void relu_add_relu(float* __restrict__ dst,
                              const float* __restrict__ res, long n) {
    long i = (long)blockIdx.x * blockDim.x + threadIdx.x;
    long st = (long)gridDim.x * blockDim.x;
    for (; i < n; i += st) {
        float a = dst[i];
        float b = res[i];
        a = a > 0.f ? a : 0.f;
        b = b > 0.f ? b : 0.f;
        dst[i] = a + b;
    }
}

// ---------------------------------------------------------------------------
extern "C" void kernel_launch(void* const* d_in, const int* in_sizes, int n_in,
                              void* d_out, int out_size, void* d_ws, size_t ws_size,
                              hipStream_t stream) {
    const float* x    = (const float*)d_in[0];   // [1, NN, 128]
    const int*   erow = (const int*)  d_in[1];   // [NEDGE]
    const int*   ecol = (const int*)  d_in[2];   // [NEDGE]
    const float* evalv= (const float*)d_in[3];   // [NEDGE]
    const float* W0   = (const float*)d_in[4];   // [128,128]
    const float* W1   = (const float*)d_in[5];   // [128,128]
    const float* W2   = (const float*)d_in[6];   // [64,128]
    float*       out  = (float*)d_out;           // [NN, 64]

    const long nh = (long)NN * DHIDF;            // 12.8M floats
    float* T  = (float*)d_ws;                    // GEMM output scratch
    float* A  = T + nh;                          // layer-0 accum (pre-ReLU h1)
    float* Bb = A + nh;                          // layer-1 accum -> h2

    const long chunks128 = (long)NEDGE * (DHIDF / 4);  // 51.2M
    const long chunks64  = (long)NEDGE * (DOUTF / 4);  // 25.6M
    const int  mblocks   = NN / 16;                    // 6250

    // ---- layer 0: A = A_hat @ (x W0^T)   (ReLU deferred/fused) ----
    gemm_wmma_f32<DINF, false><<<mblocks, 32 * (DHIDF / 16), 0, stream>>>(
        x, W0, T, DHIDF);
    zero_f32v4<<<2048, 256, 0, stream>>>((float4*)A, nh / 4);
    spmm_atomic<5, true><<<(int)((chunks128 + 255) / 256), 256, 0, stream>>>(
        erow, ecol, evalv, T, A, chunks128);

    // ---- layer 1: Bb = A_hat @ (relu(A) W1^T); Bb = relu(Bb) + relu(A) ----
    gemm_wmma_f32<DHIDF, true><<<mblocks, 32 * (DHIDF / 16), 0, stream>>>(
        A, W1, T, DHIDF);
    zero_f32v4<<<2048, 256, 0, stream>>>((float4*)Bb, nh / 4);
    spmm_atomic<5, true><<<(int)((chunks128 + 255) / 256), 256, 0, stream>>>(
        erow, ecol, evalv, T, Bb, chunks128);
    relu_add_relu<<<2048, 256, 0, stream>>>(Bb, A, nh);

    // ---- layer 2: out = A_hat @ (h2 W2^T) ----
    gemm_wmma_f32<DHIDF, false><<<mblocks, 32 * (DOUTF / 16), 0, stream>>>(
        Bb, W2, T, DOUTF);
    zero_f32v4<<<2048, 256, 0, stream>>>((float4*)out, (long)NN * DOUTF / 4);
    spmm_atomic<4, false><<<(int)((chunks64 + 255) / 256), 256, 0, stream>>>(
        erow, ecol, evalv, T, out, chunks64);
}